// DarkChannelPrior_60825326846331
// MI455X (gfx1250) — compile-verified
//
#include <hip/hip_runtime.h>

// ---------------------------------------------------------------------------
// Dark Channel Prior + guided filter for MI455X (gfx1250, wave32).
// Box blurs (R=60) are computed as banded-ones matmuls on the f32 WMMA path
// (V_WMMA_F32_16X16X4_F32): exact f32 sums, window-of-60 selection folded
// into the A (or B) operand. Memory-bound pipeline, 7 passes over 66MB planes.
// ---------------------------------------------------------------------------

#define Bn 8
#define Cn 3
#define Hh 1080
#define Ww 1920
#define ER 7          // erosion radius (KS=15)
#define PB 29         // box-blur left pad = (60-1)/2 ; right pad = 30
#define WIN 60        // box window
#define NCH 19        // ceil((60+15)/4) K=4 chunks covering a 16-tall tile's windows
#define EPSF 1e-4f
#define INV3600 (1.0f/3600.0f)

typedef float v2f __attribute__((ext_vector_type(2)));
typedef float v8f __attribute__((ext_vector_type(8)));

__device__ __forceinline__ v8f wmma4(v2f a, v2f b, v8f c) {
  // D(16x16,f32) = A(16x4,f32) x B(4x16,f32) + C
  return __builtin_amdgcn_wmma_f32_16x16x4_f32(false, a, false, b, (short)0, c,
                                               false, false);
}

__device__ __forceinline__ int reflect_idx(int r, int n) {
  if (r < 0) r = -r;               // jnp.pad 'reflect': -1 -> 1
  if (r >= n) r = 2 * n - 2 - r;   // n -> n-2
  return r;
}

__device__ __forceinline__ float selw(int m, int t) {
  // 1.0 iff input offset t lies in output-row m's 60-wide window [m, m+59]
  return (t >= m && t <= m + (WIN - 1)) ? 1.0f : 0.0f;
}

// ---------------------------------------------------------------- kernel 1 --
__global__ __launch_bounds__(256) void k_dark_min(const float* __restrict__ I,
                                                  float* __restrict__ dark) {
  int i = blockIdx.x * blockDim.x + threadIdx.x;
  const int HW = Hh * Ww;
  if (i >= Bn * HW) return;
  int b = i / HW;
  int p = i - b * HW;
  const float* base = I + (size_t)(b * Cn) * HW + p;
  float v0 = base[0];
  float v1 = base[HW];
  float v2 = base[2 * HW];
  dark[i] = fminf(v0, fminf(v1, v2));
}

// ------------------------------------------------------------ kernels 2, 3 --
__global__ __launch_bounds__(256) void k_erode_v(const float* __restrict__ in,
                                                 float* __restrict__ out) {
  int i = blockIdx.x * blockDim.x + threadIdx.x;
  const int HW = Hh * Ww;
  if (i >= Bn * HW) return;
  int b = i / HW;
  int p = i - b * HW;
  int y = p / Ww, x = p - y * Ww;
  const float* pl = in + (size_t)b * HW;
  float m = 3.4e38f;  // geodesic border: out-of-range = +inf (never wins)
  for (int dy = -ER; dy <= ER; ++dy) {
    int yy = y + dy;
    if (yy >= 0 && yy < Hh) m = fminf(m, pl[yy * Ww + x]);
  }
  out[i] = m;
}

__global__ __launch_bounds__(256) void k_erode_h(const float* __restrict__ in,
                                                 float* __restrict__ out) {
  int i = blockIdx.x * blockDim.x + threadIdx.x;
  const int HW = Hh * Ww;
  if (i >= Bn * HW) return;
  int b = i / HW;
  int p = i - b * HW;
  int y = p / Ww, x = p - y * Ww;
  const float* row = in + (size_t)b * HW + y * Ww;
  float m = 3.4e38f;
  for (int dx = -ER; dx <= ER; ++dx) {
    int xx = x + dx;
    if (xx >= 0 && xx < Ww) m = fminf(m, row[xx]);
  }
  out[i] = m;
}

// ----------------------------------------------------- WMMA vertical sums ---
// One wave per 16x16 output tile. A = 60-band selector, B = data chunk (4 rows
// x 16 cols). Four accumulators share one A: sums of d_im, dark, d_im*dark,
// d_im^2 for the guided filter.
__global__ __launch_bounds__(256) void k_vsum4(const float* __restrict__ dim_p,
                                               const float* __restrict__ drk_p,
                                               float* __restrict__ S1,
                                               float* __restrict__ S2,
                                               float* __restrict__ S3,
                                               float* __restrict__ S4,
                                               int tilesX, int tilesY, int nTiles) {
  int wave = (blockIdx.x * blockDim.x + threadIdx.x) >> 5;
  if (wave >= nTiles) return;  // wave-uniform exit: EXEC stays all-ones
  int lane = threadIdx.x & 31;
  int l15 = lane & 15;         // A: output-row M ; B/D: column N
  int kh = lane >> 4;          // K half: {0,1} vs {2,3}
  int tx = wave % tilesX;
  int t2 = wave / tilesX;
  int ty = t2 % tilesY;
  int bb = t2 / tilesY;
  int x0 = tx * 16, y0 = ty * 16;
  size_t base = (size_t)bb * Hh * Ww;

  v8f a1 = {0.f, 0.f, 0.f, 0.f, 0.f, 0.f, 0.f, 0.f};
  v8f a2 = a1, a3 = a1, a4 = a1;

  for (int c = 0; c < NCH; ++c) {
    int t0 = 4 * c + 2 * kh;
    int t1 = t0 + 1;
    int r0 = reflect_idx(y0 - PB + t0, Hh);
    int r1 = reflect_idx(y0 - PB + t1, Hh);
    float im0 = dim_p[base + (size_t)r0 * Ww + x0 + l15];
    float im1 = dim_p[base + (size_t)r1 * Ww + x0 + l15];
    float dk0 = drk_p[base + (size_t)r0 * Ww + x0 + l15];
    float dk1 = drk_p[base + (size_t)r1 * Ww + x0 + l15];
    v2f A = {selw(l15, t0), selw(l15, t1)};
    v2f B0 = {im0, im1};
    v2f B1 = {dk0, dk1};
    v2f B2 = {im0 * dk0, im1 * dk1};
    v2f B3 = {im0 * im0, im1 * im1};
    a1 = wmma4(A, B0, a1);
    a2 = wmma4(A, B1, a2);
    a3 = wmma4(A, B2, a3);
    a4 = wmma4(A, B3, a4);
  }
#pragma unroll
  for (int i = 0; i < 8; ++i) {
    int row = y0 + i + 8 * kh;   // D layout: VGPR i holds rows i / i+8
    if (row < Hh) {
      size_t o = base + (size_t)row * Ww + x0 + l15;
      S1[o] = a1[i];
      S2[o] = a2[i];
      S3[o] = a3[i];
      S4[o] = a4[i];
    }
  }
}

// ------------------------------------------- WMMA horizontal pass -> a, b ---
// Transposed formulation: D[m][n] with m = column offset, n = row offset.
// A = selector over column window, B[k][n] = S[y0+n][x0-29+k] (reflected).
__global__ __launch_bounds__(256) void k_hab(const float* __restrict__ S1,
                                             const float* __restrict__ S2,
                                             const float* __restrict__ S3,
                                             const float* __restrict__ S4,
                                             float* __restrict__ pa,
                                             float* __restrict__ pb,
                                             int tilesX, int tilesY, int nTiles) {
  int wave = (blockIdx.x * blockDim.x + threadIdx.x) >> 5;
  if (wave >= nTiles) return;
  int lane = threadIdx.x & 31;
  int l15 = lane & 15;
  int kh = lane >> 4;
  int tx = wave % tilesX;
  int t2 = wave / tilesX;
  int ty = t2 % tilesY;
  int bb = t2 / tilesY;
  int x0 = tx * 16, y0 = ty * 16;
  size_t base = (size_t)bb * Hh * Ww;

  int rowOut = y0 + l15;
  int rowRd = rowOut < Hh ? rowOut : (Hh - 1);  // clamp reads; garbage discarded
  size_t rb = base + (size_t)rowRd * Ww;

  v8f m1 = {0.f, 0.f, 0.f, 0.f, 0.f, 0.f, 0.f, 0.f};
  v8f m2 = m1, m3 = m1, m4 = m1;

  for (int c = 0; c < NCH; ++c) {
    int t0 = 4 * c + 2 * kh;
    int t1 = t0 + 1;
    int xi0 = reflect_idx(x0 - PB + t0, Ww);
    int xi1 = reflect_idx(x0 - PB + t1, Ww);
    v2f A = {selw(l15, t0), selw(l15, t1)};
    v2f B1 = {S1[rb + xi0], S1[rb + xi1]};
    v2f B2 = {S2[rb + xi0], S2[rb + xi1]};
    v2f B3 = {S3[rb + xi0], S3[rb + xi1]};
    v2f B4 = {S4[rb + xi0], S4[rb + xi1]};
    m1 = wmma4(A, B1, m1);
    m2 = wmma4(A, B2, m2);
    m3 = wmma4(A, B3, m3);
    m4 = wmma4(A, B4, m4);
  }
  if (rowOut < Hh) {
#pragma unroll
    for (int i = 0; i < 8; ++i) {
      int col = x0 + i + 8 * kh;  // transposed D: VGPR index = column offset
      float mI = m1[i] * INV3600;
      float mP = m2[i] * INV3600;
      float mIp = m3[i] * INV3600;
      float mII = m4[i] * INV3600;
      float cov = mIp - mI * mP;
      float var = mII - mI * mI;
      float a = cov / (var + EPSF);
      float b = mP - a * mI;
      size_t o = base + (size_t)rowOut * Ww + col;
      pa[o] = a;
      pb[o] = b;
    }
  }
}

// ---------------------------------------------- WMMA vertical sums of a,b ---
__global__ __launch_bounds__(256) void k_vsum2(const float* __restrict__ pa,
                                               const float* __restrict__ pb,
                                               float* __restrict__ S1,
                                               float* __restrict__ S2,
                                               int tilesX, int tilesY, int nTiles) {
  int wave = (blockIdx.x * blockDim.x + threadIdx.x) >> 5;
  if (wave >= nTiles) return;
  int lane = threadIdx.x & 31;
  int l15 = lane & 15;
  int kh = lane >> 4;
  int tx = wave % tilesX;
  int t2 = wave / tilesX;
  int ty = t2 % tilesY;
  int bb = t2 / tilesY;
  int x0 = tx * 16, y0 = ty * 16;
  size_t base = (size_t)bb * Hh * Ww;

  v8f a1 = {0.f, 0.f, 0.f, 0.f, 0.f, 0.f, 0.f, 0.f};
  v8f a2 = a1;

  for (int c = 0; c < NCH; ++c) {
    int t0 = 4 * c + 2 * kh;
    int t1 = t0 + 1;
    int r0 = reflect_idx(y0 - PB + t0, Hh);
    int r1 = reflect_idx(y0 - PB + t1, Hh);
    v2f A = {selw(l15, t0), selw(l15, t1)};
    v2f B0 = {pa[base + (size_t)r0 * Ww + x0 + l15],
              pa[base + (size_t)r1 * Ww + x0 + l15]};
    v2f B1 = {pb[base + (size_t)r0 * Ww + x0 + l15],
              pb[base + (size_t)r1 * Ww + x0 + l15]};
    a1 = wmma4(A, B0, a1);
    a2 = wmma4(A, B1, a2);
  }
#pragma unroll
  for (int i = 0; i < 8; ++i) {
    int row = y0 + i + 8 * kh;
    if (row < Hh) {
      size_t o = base + (size_t)row * Ww + x0 + l15;
      S1[o] = a1[i];
      S2[o] = a2[i];
    }
  }
}

// ----------------------------------- WMMA horizontal pass + final combine ---
__global__ __launch_bounds__(256) void k_hq(const float* __restrict__ S1,
                                            const float* __restrict__ S2,
                                            const float* __restrict__ dim_p,
                                            float* __restrict__ qout,
                                            int tilesX, int tilesY, int nTiles) {
  int wave = (blockIdx.x * blockDim.x + threadIdx.x) >> 5;
  if (wave >= nTiles) return;
  int lane = threadIdx.x & 31;
  int l15 = lane & 15;
  int kh = lane >> 4;
  int tx = wave % tilesX;
  int t2 = wave / tilesX;
  int ty = t2 % tilesY;
  int bb = t2 / tilesY;
  int x0 = tx * 16, y0 = ty * 16;
  size_t base = (size_t)bb * Hh * Ww;

  int rowOut = y0 + l15;
  int rowRd = rowOut < Hh ? rowOut : (Hh - 1);
  size_t rb = base + (size_t)rowRd * Ww;

  v8f m1 = {0.f, 0.f, 0.f, 0.f, 0.f, 0.f, 0.f, 0.f};
  v8f m2 = m1;

  for (int c = 0; c < NCH; ++c) {
    int t0 = 4 * c + 2 * kh;
    int t1 = t0 + 1;
    int xi0 = reflect_idx(x0 - PB + t0, Ww);
    int xi1 = reflect_idx(x0 - PB + t1, Ww);
    v2f A = {selw(l15, t0), selw(l15, t1)};
    v2f B1 = {S1[rb + xi0], S1[rb + xi1]};
    v2f B2 = {S2[rb + xi0], S2[rb + xi1]};
    m1 = wmma4(A, B1, m1);
    m2 = wmma4(A, B2, m2);
  }
  if (rowOut < Hh) {
#pragma unroll
    for (int i = 0; i < 8; ++i) {
      int col = x0 + i + 8 * kh;
      size_t o = base + (size_t)rowOut * Ww + col;
      float mean_a = m1[i] * INV3600;
      float mean_b = m2[i] * INV3600;
      qout[o] = mean_a * dim_p[o] + mean_b;
    }
  }
}

// ---------------------------------------------------------------------------
extern "C" void kernel_launch(void* const* d_in, const int* in_sizes, int n_in,
                              void* d_out, int out_size, void* d_ws, size_t ws_size,
                              hipStream_t stream) {
  (void)in_sizes; (void)n_in; (void)out_size; (void)ws_size;
  const float* I = (const float*)d_in[0];
  float* out = (float*)d_out;

  const size_t PS = (size_t)Bn * Hh * Ww;  // 16,588,800 floats per plane
  float* ws = (float*)d_ws;                // needs 7 planes = ~465 MB
  float* p_dark = ws + 0 * PS;
  float* p_dim  = ws + 1 * PS;
  float* p_tmp  = ws + 2 * PS;
  float* pS1    = ws + 3 * PS;
  float* pS2    = ws + 4 * PS;
  float* pS3    = ws + 5 * PS;
  float* pS4    = ws + 6 * PS;
  float* p_a = p_dark;  // dark dead after k_vsum4
  float* p_b = p_tmp;   // erosion temp dead after k_erode_h

  const int N = Bn * Hh * Ww;
  const int elemBlocks = (N + 255) / 256;

  const int tilesX = Ww / 16;              // 120
  const int tilesY = (Hh + 15) / 16;       // 68
  const int nTiles = Bn * tilesY * tilesX; // 65280
  const int tileBlocks = (nTiles + 7) / 8; // 8 waves (tiles) per 256-thread block

  k_dark_min<<<elemBlocks, 256, 0, stream>>>(I, p_dark);
  k_erode_v<<<elemBlocks, 256, 0, stream>>>(p_dark, p_tmp);
  k_erode_h<<<elemBlocks, 256, 0, stream>>>(p_tmp, p_dim);
  k_vsum4<<<tileBlocks, 256, 0, stream>>>(p_dim, p_dark, pS1, pS2, pS3, pS4,
                                          tilesX, tilesY, nTiles);
  k_hab<<<tileBlocks, 256, 0, stream>>>(pS1, pS2, pS3, pS4, p_a, p_b,
                                        tilesX, tilesY, nTiles);
  k_vsum2<<<tileBlocks, 256, 0, stream>>>(p_a, p_b, pS1, pS2,
                                          tilesX, tilesY, nTiles);
  k_hq<<<tileBlocks, 256, 0, stream>>>(pS1, pS2, p_dim, out,
                                       tilesX, tilesY, nTiles);
}